// RANSAC_65274912965345
// MI455X (gfx1250) — compile-verified
//
#include <hip/hip_runtime.h>
#include <hip/hip_bf16.h>

typedef float v2f __attribute__((ext_vector_type(2)));
typedef float v8f __attribute__((ext_vector_type(8)));

#define RANSAC_N      131072
#define RANSAC_E      1000
#define THRESH2       100.0f   // (dis <= 10)^2

// ---------------------------------------------------------------------------
// Kernel 0: zero the per-epoch inlier counters (int atomics target)
// ---------------------------------------------------------------------------
__global__ void ransac_init_kernel(int* counts, int e) {
    int i = blockIdx.x * blockDim.x + threadIdx.x;
    if (i < e) counts[i] = 0;
}

// ---------------------------------------------------------------------------
// Kernel 1: hypothesis generation — weighted (w=1) rigid fit on 5 points/epoch
// ---------------------------------------------------------------------------
__global__ void ransac_fit_kernel(const float2* __restrict__ A2,
                                  const float2* __restrict__ B2,
                                  const int* __restrict__ sidx,
                                  float* __restrict__ rotO, float* __restrict__ cO,
                                  float* __restrict__ sO,   float* __restrict__ uO,
                                  float* __restrict__ vO, int e) {
    int ep = blockIdx.x * blockDim.x + threadIdx.x;
    if (ep >= e) return;
    float2 a[5], b[5];
    float cAx = 0.f, cAy = 0.f, cBx = 0.f, cBy = 0.f;
    #pragma unroll
    for (int j = 0; j < 5; ++j) {
        int id = sidx[ep * 5 + j];
        a[j] = A2[id]; b[j] = B2[id];
        cAx += a[j].x; cAy += a[j].y; cBx += b[j].x; cBy += b[j].y;
    }
    cAx *= 0.2f; cAy *= 0.2f; cBx *= 0.2f; cBy *= 0.2f;   // ws = 5
    float sxx = 0.f, sxy = 0.f;
    #pragma unroll
    for (int j = 0; j < 5; ++j) {
        float ax = a[j].x - cAx, ay = a[j].y - cAy;
        float bx = b[j].x - cBx, by = b[j].y - cBy;
        sxx += ax * bx + ay * by;
        sxy += ax * by - ay * bx;
    }
    float r = atan2f(sxy, sxx);
    float cr = cosf(r), sr = sinf(r);
    rotO[ep] = r;
    cO[ep] = cr; sO[ep] = sr;
    uO[ep] = cBx - (cr * cAx - sr * cAy);
    vO[ep] = cBy - (sr * cAx + cr * cAy);
}

// ---------------------------------------------------------------------------
// Kernel 2: scoring via V_WMMA_F32_16X16X4_F32.
//   A (16x4):  row 2k   = [c_k, -s_k, u_k, 0]   (transformed x of epoch k)
//              row 2k+1 = [s_k,  c_k, v_k, 0]   (transformed y of epoch k)
//   B (4x16):  col n    = [xA_n, yA_n, 1, 0]
//   D[2k][n]   = c*xA - s*yA + u ;  D[2k+1][n] = s*xA + c*yA + v
//   errx = xB - D[2k] ; erry = yB - D[2k+1]  -> lane-local (adjacent VGPRs).
//   One WMMA covers 8 epochs x 16 points. Wave: 8 epochs x 2048 points.
// ---------------------------------------------------------------------------
__global__ __launch_bounds__(256) void ransac_score_kernel(
        const float2* __restrict__ A2, const float2* __restrict__ B2,
        const float* __restrict__ cP, const float* __restrict__ sP,
        const float* __restrict__ uP, const float* __restrict__ vP,
        int* __restrict__ counts) {
    const int lane = threadIdx.x & 31;
    const int wave = threadIdx.x >> 5;
    const int eg   = blockIdx.y;                 // epoch group: 8 epochs
    const int base = (blockIdx.x * 8 + wave) * 2048;

    const int  m   = lane & 15;                  // A-matrix row this lane feeds
    const int  el  = m >> 1;                     // epoch-in-group of that row
    const bool odd = (m & 1) != 0;               // y-row vs x-row
    const int  ep  = eg * 8 + el;
    const float cc = cP[ep], ss = sP[ep], uu = uP[ep], vv = vP[ep];

    // 16x4 f32 A layout: lanes 0-15 -> K=0,1 ; lanes 16-31 -> K=2,3 (row m)
    v2f afrag;
    afrag.x = (lane < 16) ? (odd ? ss : cc) : (odd ? vv : uu);
    afrag.y = (lane < 16) ? (odd ? cc : -ss) : 0.0f;

    int cnt0 = 0, cnt1 = 0, cnt2 = 0, cnt3 = 0;
    for (int it = 0; it < 128; ++it) {
        const int p = base + it * 16 + m;
        const float2 pa = A2[p];
        const float2 pb = B2[p];
        // 4x16 f32 B layout: lanes 0-15 -> K=0,1 ; lanes 16-31 -> K=2,3 (col m)
        v2f bfrag;
        bfrag.x = (lane < 16) ? pa.x : 1.0f;
        bfrag.y = (lane < 16) ? pa.y : 0.0f;
        v8f czero = {};
        v8f d = __builtin_amdgcn_wmma_f32_16x16x4_f32(
            false, afrag, false, bfrag, (short)0, czero, false, false);
        // lanes 0-15 see epochs eg*8+0..3 ; lanes 16-31 see eg*8+4..7
        float ex, ey, d2;
        ex = pb.x - d[0]; ey = pb.y - d[1]; d2 = ex*ex + ey*ey; cnt0 += (d2 <= THRESH2);
        ex = pb.x - d[2]; ey = pb.y - d[3]; d2 = ex*ex + ey*ey; cnt1 += (d2 <= THRESH2);
        ex = pb.x - d[4]; ey = pb.y - d[5]; d2 = ex*ex + ey*ey; cnt2 += (d2 <= THRESH2);
        ex = pb.x - d[6]; ey = pb.y - d[7]; d2 = ex*ex + ey*ey; cnt3 += (d2 <= THRESH2);
    }
    // butterfly-reduce within each 16-lane half (stays inside the half)
    #pragma unroll
    for (int off = 8; off; off >>= 1) {
        cnt0 += __shfl_xor(cnt0, off);
        cnt1 += __shfl_xor(cnt1, off);
        cnt2 += __shfl_xor(cnt2, off);
        cnt3 += __shfl_xor(cnt3, off);
    }
    if ((lane & 15) == 0) {
        const int ebase = eg * 8 + ((lane >> 4) << 2);
        atomicAdd(&counts[ebase + 0], cnt0);
        atomicAdd(&counts[ebase + 1], cnt1);
        atomicAdd(&counts[ebase + 2], cnt2);
        atomicAdd(&counts[ebase + 3], cnt3);
    }
}

// ---------------------------------------------------------------------------
// Kernel 3: argmax over counts (first max wins, like jnp.argmax)
// ---------------------------------------------------------------------------
__global__ void ransac_argmax_kernel(const int* __restrict__ counts,
                                     int* __restrict__ best, int e) {
    __shared__ int sc[1024];
    __shared__ int si[1024];
    int t = threadIdx.x;
    int cb = -1, ib = 0;
    for (int i = t; i < e; i += 1024) {
        int cv = counts[i];
        if (cv > cb || (cv == cb && i < ib)) { cb = cv; ib = i; }
    }
    sc[t] = cb; si[t] = ib;
    __syncthreads();
    for (int s = 512; s; s >>= 1) {
        if (t < s) {
            if (sc[t + s] > sc[t] || (sc[t + s] == sc[t] && si[t + s] < si[t])) {
                sc[t] = sc[t + s]; si[t] = si[t + s];
            }
        }
        __syncthreads();
    }
    if (t == 0) *best = si[0];
}

// ---------------------------------------------------------------------------
// Kernel 4: weighted-refit partial sums over all N points (deterministic:
// per-block partials to fixed slots, no float atomics)
//   acc = { ws, S(w*Ax), S(w*Ay), S(w*Bx), S(w*By),
//           S(w*(AxBx+AyBy)), S(w*(AxBy-AyBx)) }
// ---------------------------------------------------------------------------
__global__ __launch_bounds__(256) void ransac_refit_partial_kernel(
        const float2* __restrict__ A2, const float2* __restrict__ B2,
        const float* __restrict__ cP, const float* __restrict__ sP,
        const float* __restrict__ uP, const float* __restrict__ vP,
        const int* __restrict__ best, float* __restrict__ partials, int n) {
    const int b = *best;
    const float cc = cP[b], ss = sP[b], uu = uP[b], vv = vP[b];
    float acc[7] = {0.f, 0.f, 0.f, 0.f, 0.f, 0.f, 0.f};
    for (int i = blockIdx.x * blockDim.x + threadIdx.x; i < n;
         i += gridDim.x * blockDim.x) {
        float2 pa = A2[i], pb = B2[i];
        float ex = pb.x - (cc * pa.x - ss * pa.y + uu);
        float ey = pb.y - (ss * pa.x + cc * pa.y + vv);
        float w = (ex * ex + ey * ey <= THRESH2) ? 1.0f : 0.0f;
        acc[0] += w;
        acc[1] += w * pa.x; acc[2] += w * pa.y;
        acc[3] += w * pb.x; acc[4] += w * pb.y;
        acc[5] += w * (pa.x * pb.x + pa.y * pb.y);
        acc[6] += w * (pa.x * pb.y - pa.y * pb.x);
    }
    // wave butterfly (fixed order -> deterministic)
    #pragma unroll
    for (int off = 16; off; off >>= 1)
        #pragma unroll
        for (int j = 0; j < 7; ++j) acc[j] += __shfl_xor(acc[j], off);
    __shared__ float sh[8][8];
    int lane = threadIdx.x & 31, wv = threadIdx.x >> 5;
    if (lane == 0)
        for (int j = 0; j < 7; ++j) sh[wv][j] = acc[j];
    __syncthreads();
    if (threadIdx.x == 0) {
        float tot[7] = {0.f, 0.f, 0.f, 0.f, 0.f, 0.f, 0.f};
        for (int w2 = 0; w2 < 8; ++w2)
            for (int j = 0; j < 7; ++j) tot[j] += sh[w2][j];
        for (int j = 0; j < 7; ++j) partials[blockIdx.x * 8 + j] = tot[j];
    }
}

// ---------------------------------------------------------------------------
// Kernel 5: fixed-order reduce of partials, final fit, write [2,3,3] output
// ---------------------------------------------------------------------------
__global__ void ransac_finalize_kernel(const float* __restrict__ partials, int nblk,
                                       const float* __restrict__ cP,
                                       const float* __restrict__ sP,
                                       const float* __restrict__ uP,
                                       const float* __restrict__ vP,
                                       const int* __restrict__ best,
                                       float* __restrict__ out) {
    if (threadIdx.x != 0 || blockIdx.x != 0) return;
    float acc[7] = {0.f, 0.f, 0.f, 0.f, 0.f, 0.f, 0.f};
    for (int bk = 0; bk < nblk; ++bk)
        for (int j = 0; j < 7; ++j) acc[j] += partials[bk * 8 + j];
    float ws  = fmaxf(acc[0], 1e-8f);
    float cAx = acc[1] / ws, cAy = acc[2] / ws;
    float cBx = acc[3] / ws, cBy = acc[4] / ws;
    // centered cross-sums via single-pass expansion
    float sxx = acc[5] - (acc[1] * acc[3] + acc[2] * acc[4]) / ws;
    float sxy = acc[6] - (acc[1] * acc[4] - acc[2] * acc[3]) / ws;
    float r  = atan2f(sxy, sxx);
    float cr = cosf(r), sr = sinf(r);
    float uf = cBx - (cr * cAx - sr * cAy);
    float vf = cBy - (sr * cAx + cr * cAy);
    // E_final
    out[0] = cr;  out[1] = -sr; out[2] = uf;
    out[3] = sr;  out[4] = cr;  out[5] = vf;
    out[6] = 0.f; out[7] = 0.f; out[8] = 1.f;
    // E_best
    int b = *best;
    float cb = cP[b], sb = sP[b];
    out[9]  = cb;  out[10] = -sb; out[11] = uP[b];
    out[12] = sb;  out[13] = cb;  out[14] = vP[b];
    out[15] = 0.f; out[16] = 0.f; out[17] = 1.f;
}

// ---------------------------------------------------------------------------
extern "C" void kernel_launch(void* const* d_in, const int* in_sizes, int n_in,
                              void* d_out, int out_size, void* d_ws, size_t ws_size,
                              hipStream_t stream) {
    (void)in_sizes; (void)n_in; (void)out_size; (void)ws_size;
    const float2* A2  = (const float2*)d_in[0];   // pstA [1,N,2] f32
    const float2* B2  = (const float2*)d_in[1];   // pstB [1,N,2] f32
    const int*   sidx = (const int*)d_in[2];      // subset_idx [E,5] i32
    float* out = (float*)d_out;                   // [2,3,3] f32

    // workspace layout (floats): rot|c|s|u|v @1024 stride, counts, best, partials
    float* f        = (float*)d_ws;
    float* rot      = f;
    float* cP       = f + 1024;
    float* sP       = f + 2048;
    float* uP       = f + 3072;
    float* vP       = f + 4096;
    int*   counts   = (int*)(f + 5120);           // [E] int
    int*   best     = (int*)(f + 6144);           // [1] int
    float* partials = f + 6152;                   // [256*8] f32

    ransac_init_kernel<<<4, 256, 0, stream>>>(counts, RANSAC_E);
    ransac_fit_kernel<<<(RANSAC_E + 255) / 256, 256, 0, stream>>>(
        A2, B2, sidx, rot, cP, sP, uP, vP, RANSAC_E);
    // 8 point-chunk blocks (x 8 waves = 64 chunks of 2048 pts) x 125 epoch groups
    ransac_score_kernel<<<dim3(8, 125), 256, 0, stream>>>(
        A2, B2, cP, sP, uP, vP, counts);
    ransac_argmax_kernel<<<1, 1024, 0, stream>>>(counts, best, RANSAC_E);
    ransac_refit_partial_kernel<<<256, 256, 0, stream>>>(
        A2, B2, cP, sP, uP, vP, best, partials, RANSAC_N);
    ransac_finalize_kernel<<<1, 64, 0, stream>>>(
        partials, 256, cP, sP, uP, vP, best, out);
}